// NPTransitionPrior_9268539425199
// MI455X (gfx1250) — compile-verified
//
#include <hip/hip_runtime.h>
#include <math.h>

// ---------------------------------------------------------------------------
// NPTransitionPrior forward on gfx1250 (CDNA5 / MI455X), fp32 WMMA path.
//   D=8 per-coordinate MLPs: (N,9)->64->64->64->1 + Jacobian-diagonal stream.
//   V_WMMA_F32_16X16X4_F32, wave32; each wave owns 16 samples.
// ---------------------------------------------------------------------------

typedef float v2f __attribute__((ext_vector_type(2)));
typedef float v8f __attribute__((ext_vector_type(8)));

#define DD      8
#define HH      64
#define IN_DIM  9
#define BB      128
#define LL      256
#define NTOT    (BB * LL)        // 32768 samples
#define SLOPE   0.2f
#define AST     66               // padded row stride of per-wave act buffers
#define WAVES   8
#define ROWSBLK (WAVES * 16)     // 128 samples per block

static __device__ __forceinline__ v8f wmma_f32(v2f a, v2f b, v8f c) {
  // D(16x16 f32) = A(16x4 f32) * B(4x16 f32) + C
  return __builtin_amdgcn_wmma_f32_16x16x4_f32(
      /*neg_a=*/false, a, /*neg_b=*/false, b,
      /*c_mod=*/(short)0, c, /*reuse_a=*/false, /*reuse_b=*/false);
}

__global__ __launch_bounds__(256) void flow_mlp_kernel(
    const float* __restrict__ x,
    const float* __restrict__ W1, const float* __restrict__ b1,
    const float* __restrict__ W2, const float* __restrict__ b2,
    const float* __restrict__ W3, const float* __restrict__ b3,
    const float* __restrict__ Wo, const float* __restrict__ bo,
    float* __restrict__ out, float* __restrict__ ws)
{
  // ---- LDS: weights (shared by block) + per-wave activation buffers -------
  __shared__ float sW1T[16 * 64];          // [k][h], k 0..15 (9..15 zero)
  __shared__ float sW2T[64 * 64];          // [k=h_in][g=h_out]
  __shared__ float sW3T[64 * 64];
  __shared__ float sB1[64], sB2[64], sB3[64], sWoL[64];
  __shared__ float sBufA[WAVES][16 * AST]; // activation stream, [sample][feat]
  __shared__ float sBufD[WAVES][16 * AST]; // derivative stream

  const int tid = threadIdx.x;
  const int d   = blockIdx.y;

  // ---- stage weights (transposed so B-fragment loads are along K) ---------
  for (int i = tid; i < 16 * 64; i += 256) {
    const int k = i >> 6, h = i & 63;
    sW1T[i] = (k < IN_DIM) ? W1[(d * HH + h) * IN_DIM + k] : 0.f;
  }
  for (int i = tid; i < 64 * 64; i += 256) {
    const int g = i >> 6, h = i & 63;
    sW2T[h * 64 + g] = W2[d * 4096 + i];   // W2[d][g][h] -> [h][g]
    sW3T[h * 64 + g] = W3[d * 4096 + i];
  }
  if (tid < 64) {
    sB1[tid]  = b1[d * HH + tid];
    sB2[tid]  = b2[d * HH + tid];
    sB3[tid]  = b3[d * HH + tid];
    sWoL[tid] = Wo[d * HH + tid];
  }
  __syncthreads();

  const int wave = tid >> 5;
  const int lane = tid & 31;
  const int lx   = lane & 15;
  const int half = lane >> 4;      // 0: lanes 0-15, 1: lanes 16-31
  const int koff = half * 2;       // A/B fragment K offset per ISA layout

  float* sA = sBufA[wave];
  float* sD = sBufD[wave];

  // ---- stage inputs: 16 samples per wave, inp = [x_lag(8) | x_t[d] | 0..] -
  const int n0 = blockIdx.x * ROWSBLK + wave * 16;   // 16 | 256 so no b-wrap
  const int bb = n0 >> 8;
  const int l0 = n0 & 255;
  {
    const int s = lane >> 1, part = lane & 1;        // float4 per lane
    const float4 v = *reinterpret_cast<const float4*>(
        x + (size_t)(bb * 257 + l0 + s) * 8 + part * 4);
    float* dst = &sA[s * AST + part * 4];
    dst[0] = v.x; dst[1] = v.y; dst[2] = v.z; dst[3] = v.w;
    if (lane < 16) {
      sA[lane * AST + 8] = x[(size_t)(bb * 257 + l0 + lane + 1) * 8 + d];
    } else {
      sA[lx * AST + 9]  = 0.f;
      sA[lx * AST + 10] = 0.f;
      sA[lx * AST + 11] = 0.f;
    }
  }

  // ---- layer 1: (16 x 12pad) x (12pad x 64), 3 K-steps, 4 N-tiles ---------
  v8f acc1[4];
  {
    v2f af[3];
#pragma unroll
    for (int kk = 0; kk < 3; ++kk) {
      const int k = kk * 4 + koff;
      af[kk].x = sA[lx * AST + k];
      af[kk].y = sA[lx * AST + k + 1];
    }
#pragma unroll
    for (int t = 0; t < 4; ++t) {
      v8f c = {0.f, 0.f, 0.f, 0.f, 0.f, 0.f, 0.f, 0.f};
#pragma unroll
      for (int kk = 0; kk < 3; ++kk) {
        const int k = kk * 4 + koff;
        v2f b;
        b.x = sW1T[k * 64 + t * 16 + lx];
        b.y = sW1T[(k + 1) * 64 + t * 16 + lx];
        c = wmma_f32(af[kk], b, c);
      }
      acc1[t] = c;
    }
  }
  // epilogue: leaky act; d1 = m1 * W1[:, :, 8]
#pragma unroll
  for (int t = 0; t < 4; ++t) {
    const int f   = t * 16 + lx;
    const float w8 = sW1T[8 * 64 + f];
    const float bv = sB1[f];
#pragma unroll
    for (int v = 0; v < 8; ++v) {
      const float h = acc1[t][v] + bv;
      const float m = (h >= 0.f) ? 1.f : SLOPE;
      const int  ms = v + 8 * half;
      sA[ms * AST + f] = m * h;
      sD[ms * AST + f] = m * w8;
    }
  }

  // ---- layers 2 & 3: dual-stream (16x64)x(64x64), 16 K-steps, 4 N-tiles ---
  auto layer = [&](const float* __restrict__ WT, const float* __restrict__ bias) {
    v2f af[16], df[16];
#pragma unroll
    for (int kk = 0; kk < 16; ++kk) {
      const int k = kk * 4 + koff;
      af[kk].x = sA[lx * AST + k];  af[kk].y = sA[lx * AST + k + 1];
      df[kk].x = sD[lx * AST + k];  df[kk].y = sD[lx * AST + k + 1];
    }
#pragma unroll
    for (int t = 0; t < 4; ++t) {
      v8f ca = {0.f, 0.f, 0.f, 0.f, 0.f, 0.f, 0.f, 0.f};
      v8f cd = {0.f, 0.f, 0.f, 0.f, 0.f, 0.f, 0.f, 0.f};
#pragma unroll
      for (int kk = 0; kk < 16; ++kk) {
        const int k = kk * 4 + koff;
        v2f b;
        b.x = WT[k * 64 + t * 16 + lx];
        b.y = WT[(k + 1) * 64 + t * 16 + lx];
        ca = wmma_f32(af[kk], b, ca);   // activation stream
        cd = wmma_f32(df[kk], b, cd);   // derivative stream (same B)
      }
      const int f   = t * 16 + lx;
      const float bv = bias[f];
#pragma unroll
      for (int v = 0; v < 8; ++v) {
        const float h = ca[v] + bv;
        const float m = (h >= 0.f) ? 1.f : SLOPE;
        const int  ms = v + 8 * half;
        sA[ms * AST + f] = m * h;       // a_next (frags already in regs)
        sD[ms * AST + f] = m * cd[v];   // d_next
      }
    }
  };
  layer(sW2T, sB2);
  layer(sW3T, sB3);

  // ---- output layer: 64->1 dot; lanes 0-15 residual, 16-31 dlast ----------
  {
    const float* buf = (lane < 16) ? sA : sD;
    float s = 0.f;
#pragma unroll 8
    for (int k = 0; k < 64; ++k) s += buf[lx * AST + k] * sWoL[k];
    const int n = n0 + lx;
    if (lane < 16) {
      out[(size_t)n * DD + d] = s + bo[d];               // residuals[b][l][d]
    } else {
      ws[(size_t)d * NTOT + n] = s;                      // raw dlast
    }
  }
}

// Sum log|dlast| over d (deterministic reduction, no atomics).
__global__ __launch_bounds__(256) void logdet_kernel(
    const float* __restrict__ ws, float* __restrict__ out)
{
  const int r = blockIdx.x * 256 + threadIdx.x;
  if (r >= NTOT) return;
  float s = 0.f;
#pragma unroll
  for (int d = 0; d < DD; ++d) s += logf(fabsf(ws[(size_t)d * NTOT + r]));
  out[(size_t)NTOT * DD + r] = s;
}

extern "C" void kernel_launch(void* const* d_in, const int* in_sizes, int n_in,
                              void* d_out, int out_size, void* d_ws, size_t ws_size,
                              hipStream_t stream) {
  const float* x  = (const float*)d_in[0];
  const float* W1 = (const float*)d_in[1];
  const float* b1 = (const float*)d_in[2];
  const float* W2 = (const float*)d_in[3];
  const float* b2 = (const float*)d_in[4];
  const float* W3 = (const float*)d_in[5];
  const float* b3 = (const float*)d_in[6];
  const float* Wo = (const float*)d_in[7];
  const float* bo = (const float*)d_in[8];
  float* out = (float*)d_out;
  float* ws  = (float*)d_ws;   // needs D*N*4 = 1 MB

  dim3 grid(NTOT / ROWSBLK, DD);   // 256 x 8 blocks
  flow_mlp_kernel<<<grid, dim3(256), 0, stream>>>(
      x, W1, b1, W2, b2, W3, b3, Wo, bo, out, ws);
  logdet_kernel<<<dim3(NTOT / 256), dim3(256), 0, stream>>>(ws, out);
}